// CrossCorrelation_38405597561347
// MI455X (gfx1250) — compile-verified
//
#include <hip/hip_runtime.h>

typedef __attribute__((ext_vector_type(2))) float v2f;
typedef __attribute__((ext_vector_type(8))) float v8f;

#define B_SAMPLES 64

// ---- compile-time for-loop: guarantees full unroll + constexpr indices ----
template<int V> struct Int { static constexpr int value = V; };

template<int LO, int HI, typename F>
__device__ __forceinline__ void static_for(F f) {
    if constexpr (LO < HI) {
        f(Int<LO>{});
        static_for<LO + 1, HI>(f);
    }
}

// ---------------------------------------------------------------------------
// Correlation: out[y,x] = sum_{i,j} s[y+i, x+j] * t[T-1-i, T-1-j]
// One wave computes a 16(w) x 32(h) output tile via V_WMMA_F32_16X16X4_F32:
//   per staged input row p:  r_p[i,v] = sum_j A[i,j] * B[j,v]
//     A = flipped template (16x16 zero-padded, in regs), B = Hankel of row p
// Diagonal reduction uses a split-accumulator layout so every contribution
// is a full-wave v_add_f32 (no shuffles/cndmask in the loop):
//   acc[a] (a in [-8,32)): low lanes = row a+8 partial over i in [0,8),
//                          high lanes = row a  partial over i in [8,16).
//   For e = q-8-a in [0,8): acc[a] += r[e] covers both halves usefully.
// Final combine (16x: 2 cndmask + 1 shfl + 1 add) assembles row totals.
// Staging uses global_load_async_to_lds_b128 (ASYNCcnt).
// Workgroup = 8 waves = 128 output columns, 32 output rows.
// ---------------------------------------------------------------------------
template<int H, int T, int N, int P>
__global__ __launch_bounds__(256) void corr_wmma_kernel(
    const float* __restrict__ s, const float* __restrict__ tpl,
    float* __restrict__ corr)
{
    constexpr int ROWS   = T + 31;        // input rows per 32-row output block
    constexpr int SPAN   = 144;           // 128 cols + 15 halo, 16B multiple
    constexpr int NCHUNK = (T + 3) / 4;   // K chunks of 4 (template width)

    __shared__ float lds_s[ROWS][SPAN];
    __shared__ float lds_t[16][16];

    const int tid  = threadIdx.x;
    const int lane = tid & 31;
    const int wave = tid >> 5;
    const int b    = blockIdx.z;
    const int y0   = blockIdx.y * 32;
    const int x0w  = blockIdx.x * 128;

    // Stage flipped + zero-padded template (A matrix source).
    {
        int r = tid >> 4, c = tid & 15;
        float tv = 0.0f;
        if (r < T && c < T)
            tv = tpl[((size_t)b * T + (T - 1 - r)) * T + (T - 1 - c)];
        lds_t[r][c] = tv;
    }

    // Stage input rows via async copy, 16B per lane-op, clamped
    // (clamped cells only feed discarded outputs).
    {
        const float* sbase = s + (size_t)b * H * H;
        for (int idx = tid; idx < ROWS * (SPAN / 4); idx += 256) {
            int rr = idx / (SPAN / 4);
            int c4 = (idx - rr * (SPAN / 4)) * 4;
            int gy = y0 + rr;  if (gy > H - 1) gy = H - 1;
            int gx = x0w + c4; if (gx > H - 4) gx = H - 4;
            unsigned ldsoff = (unsigned)(uintptr_t)&lds_s[rr][c4];
            unsigned voff   = (unsigned)(((gy * H) + gx) * 4);
            asm volatile("global_load_async_to_lds_b128 %0, %1, %2"
                         :: "v"(ldsoff), "v"(voff), "s"(sbase) : "memory");
        }
        asm volatile("s_wait_asynccnt 0x0" ::: "memory");
    }
    __syncthreads();

    // A / B operand lane mapping (f32 16x16x4):
    //   lanes 0-15 hold K = {0,1} in the 2 VGPRs, lanes 16-31 hold K = {2,3};
    //   m (A row) / n (B col) = lane & 15.
    const int m  = lane & 15;
    const int kb = (lane >> 4) << 1;

    v2f Amat[NCHUNK];
    static_for<0, NCHUNK>([&](auto cc) {
        constexpr int c = decltype(cc)::value;
        Amat[c].x = lds_t[m][4 * c + kb];
        Amat[c].y = lds_t[m][4 * c + kb + 1];
    });

    // Split accumulators: accArr[a+8] is acc[a], a in [-8, 32).
    float accArr[40];
    static_for<0, 40>([&](auto ii) { accArr[decltype(ii)::value] = 0.0f; });

    const int xloc = (wave << 4) + m;   // column base within the LDS span

    static_for<0, ROWS>([&](auto qc) {
        constexpr int q = decltype(qc)::value;
        const float* row = &lds_s[q][xloc + kb];

        // B operands for this row: distinct temporaries, constant LDS offsets.
        v2f Bv[NCHUNK];
        static_for<0, NCHUNK>([&](auto cc) {
            constexpr int c = decltype(cc)::value;
            Bv[c].x = row[4 * c];
            Bv[c].y = row[4 * c + 1];
        });

        v8f r = {0.f, 0.f, 0.f, 0.f, 0.f, 0.f, 0.f, 0.f};
        static_for<0, NCHUNK>([&](auto cc) {
            constexpr int c = decltype(cc)::value;
            // D = A x B + C   (chained over K chunks)
            r = __builtin_amdgcn_wmma_f32_16x16x4_f32(
                    false, Amat[c], false, Bv[c], (short)0, r, false, false);
        });

        // Full-wave diagonal accumulation: acc[a] += r[e], a = q-8-e.
        static_for<0, 8>([&](auto ec) {
            constexpr int e = decltype(ec)::value;
            constexpr int a = q - 8 - e;
            if constexpr (a >= -8 && a < 32)
                accArr[a + 8] += r[e];
        });
    });

    // Combine halves and store: row k total (lanes 0-15) paired with
    // row k+16 total (lanes 16-31).
    const int x = x0w + (wave << 4) + m;
    const bool lo = (lane < 16);
    if (x < N) {
        static_for<0, 16>([&](auto kc) {
            constexpr int k = decltype(kc)::value;
            // P : low = lowpart[row k]        (acc[k-8].low  -> accArr[k])
            //     high= highpart[row k+16]    (acc[k+16].high-> accArr[k+24])
            float Pv = lo ? accArr[k] : accArr[k + 24];
            // Q : low = acc[k+8].low = lowpart[row k+16]  (accArr[k+16])
            //     high= acc[k].high  = highpart[row k]    (accArr[k+8])
            float Qv = lo ? accArr[k + 16] : accArr[k + 8];
            float Qs = __shfl_xor(Qv, 16, 32);  // swap halves into place
            float tot = Pv + Qs;
            int y = y0 + k + (lo ? 0 : 16);
            if (y < N)
                corr[((size_t)b * N + y) * P + x] = tot;
        });
    }
}

// ---------------------------------------------------------------------------
// jax.image.resize(method='bilinear', antialias=True) from NxN -> 128x128.
// Triangle kernel, kernel_scale = max(N/128, 1), edge-renormalized weights.
// Writes channel `chan` of the [B,128,128,3] output.
// ---------------------------------------------------------------------------
template<int N, int P>
__global__ __launch_bounds__(256) void resize_aa_kernel(
    const float* __restrict__ corr, float* __restrict__ out, int chan)
{
    int idx = blockIdx.x * 256 + threadIdx.x;
    if (idx >= B_SAMPLES * 128 * 128) return;
    int ox = idx & 127;
    int oy = (idx >> 7) & 127;
    int b  = idx >> 14;

    const float inv_scale = (float)N / 128.0f;
    const float ks = inv_scale > 1.0f ? inv_scale : 1.0f;

    float cy = ((float)oy + 0.5f) * inv_scale - 0.5f;
    float cx = ((float)ox + 0.5f) * inv_scale - 0.5f;

    int ylo = (int)ceilf(cy - ks);  if (ylo < 0) ylo = 0;
    int yhi = (int)floorf(cy + ks); if (yhi > N - 1) yhi = N - 1;
    int xlo = (int)ceilf(cx - ks);  if (xlo < 0) xlo = 0;
    int xhi = (int)floorf(cx + ks); if (xhi > N - 1) xhi = N - 1;

    int nx = xhi - xlo + 1; if (nx > 9) nx = 9;
    int ny = yhi - ylo + 1; if (ny > 9) ny = 9;

    float wx[9]; float sx = 0.0f;
#pragma unroll
    for (int k = 0; k < 9; ++k) {
        float w = 0.0f;
        if (k < nx) {
            float d = fabsf((float)(xlo + k) - cx) / ks;
            w = 1.0f - d; if (w < 0.0f) w = 0.0f;
        }
        wx[k] = w; sx += w;
    }

    const float* base = corr + (size_t)b * N * P;
    float accum = 0.0f, sy = 0.0f;
    for (int kyi = 0; kyi < 9; ++kyi) {
        if (kyi >= ny) break;
        float dy = fabsf((float)(ylo + kyi) - cy) / ks;
        float wy = 1.0f - dy; if (wy < 0.0f) wy = 0.0f;
        sy += wy;
        if (wy != 0.0f) {
            const float* rowp = base + (size_t)(ylo + kyi) * P + xlo;
            float rowacc = 0.0f;
#pragma unroll
            for (int kx = 0; kx < 9; ++kx)
                if (kx < nx) rowacc += wx[kx] * rowp[kx];
            accum += wy * rowacc;
        }
    }
    accum /= (sx * sy);
    out[(((size_t)b * 128 + oy) * 128 + ox) * 3 + chan] = accum;
}

// ---------------------------------------------------------------------------
extern "C" void kernel_launch(void* const* d_in, const int* in_sizes, int n_in,
                              void* d_out, int out_size, void* d_ws, size_t ws_size,
                              hipStream_t stream)
{
    (void)in_sizes; (void)n_in; (void)out_size; (void)ws_size;

    const float* s1 = (const float*)d_in[0];
    const float* s2 = (const float*)d_in[1];
    const float* s3 = (const float*)d_in[2];
    const float* t1 = (const float*)d_in[3];
    const float* t2 = (const float*)d_in[4];
    const float* t3 = (const float*)d_in[5];
    float* out  = (float*)d_out;
    float* corr = (float*)d_ws;   // reused scale-by-scale (max 64*497*512*4 B)

    const int npix = B_SAMPLES * 128 * 128;
    dim3 rgrd((npix + 255) / 256), rblk(256);

    // scale 1: 512 -> corr 497 (template 16)
    corr_wmma_kernel<512, 16, 497, 512><<<dim3(4, 16, 64), 256, 0, stream>>>(s1, t1, corr);
    resize_aa_kernel<497, 512><<<rgrd, rblk, 0, stream>>>(corr, out, 0);

    // scale 2: 256 -> corr 249 (template 8, zero-padded to 16)
    corr_wmma_kernel<256, 8, 249, 256><<<dim3(2, 8, 64), 256, 0, stream>>>(s2, t2, corr);
    resize_aa_kernel<249, 256><<<rgrd, rblk, 0, stream>>>(corr, out, 1);

    // scale 3: 128 -> corr 125 (template 4, zero-padded to 16)
    corr_wmma_kernel<128, 4, 125, 128><<<dim3(1, 4, 64), 256, 0, stream>>>(s3, t3, corr);
    resize_aa_kernel<125, 128><<<rgrd, rblk, 0, stream>>>(corr, out, 2);
}